// TransELayer_23021024706783
// MI455X (gfx1250) — compile-verified
//
#include <hip/hip_runtime.h>
#include <cstdint>

#define N_NODES 10000
#define N_RELS  500
#define HDIM    128
#define LQ      128

#define Q_TILE 32
#define E_TILE 64
#define LDS_STRIDE 132   // 132 mod 64 = 4 banks/row -> conflict-free b64 reads

typedef float v2f __attribute__((ext_vector_type(2)));
typedef float v4f __attribute__((ext_vector_type(4)));
typedef float v8f __attribute__((ext_vector_type(8)));

// Match the builtin's parameter type exactly: vector_size(16) int
typedef int i4v __attribute__((vector_size(16)));
typedef __attribute__((address_space(1))) i4v* as1_i4p;
typedef __attribute__((address_space(3))) i4v* as3_i4p;

__device__ __forceinline__ float wave_reduce_add(float v) {
    #pragma unroll
    for (int off = 16; off > 0; off >>= 1) v += __shfl_xor(v, off, 32);
    return v;
}

// ---- CDNA5 async global->LDS copy (16B per lane) --------------------------
__device__ __forceinline__ void async_cp16(const float* g, float* l) {
#if __has_builtin(__builtin_amdgcn_global_load_async_to_lds_b128)
    __builtin_amdgcn_global_load_async_to_lds_b128(
        (as1_i4p)(uintptr_t)g,
        (as3_i4p)(unsigned)(uintptr_t)l,
        0, 0);
#else
    *(v4f*)l = *(const v4f*)g;   // synchronous fallback
#endif
}

__device__ __forceinline__ void async_wait_all() {
#if __has_builtin(__builtin_amdgcn_global_load_async_to_lds_b128)
#  if __has_builtin(__builtin_amdgcn_s_wait_asynccnt)
    __builtin_amdgcn_s_wait_asynccnt(0);
#  else
    asm volatile("s_wait_asynccnt 0" ::: "memory");
#  endif
#endif
}

// ---- Kernel 1: L2-normalize all node embeddings --------------------------
__global__ void __launch_bounds__(256) k_normalize_nodes(const float* __restrict__ node,
                                                         float* __restrict__ out) {
    const int wave = threadIdx.x >> 5;
    const int lane = threadIdx.x & 31;
    const int row  = blockIdx.x * 8 + wave;
    if (row >= N_NODES) return;
    v4f x = ((const v4f*)(node + (size_t)row * HDIM))[lane];
    float ss = x.x * x.x + x.y * x.y + x.z * x.z + x.w * x.w;
    ss = wave_reduce_add(ss);
    const float inv = 1.0f / fmaxf(__builtin_sqrtf(ss), 1e-12f);
    ((v4f*)(out + (size_t)row * HDIM))[lane] = x * inv;
}

// ---- Kernel 2: query[l] = norm(node[ent_idx[l]]) + norm(rel[rel_idx[l]]) --
__global__ void __launch_bounds__(256) k_build_query(const float* __restrict__ ent_norm,
                                                     const float* __restrict__ rel,
                                                     const int* __restrict__ ent_idx,
                                                     const int* __restrict__ rel_idx,
                                                     float* __restrict__ query) {
    const int wave = threadIdx.x >> 5;
    const int lane = threadIdx.x & 31;
    const int l = blockIdx.x * 8 + wave;
    if (l >= LQ) return;
    const int ei = ent_idx[l];
    const int ri = rel_idx[l];
    v4f e = ((const v4f*)(ent_norm + (size_t)ei * HDIM))[lane];   // already normalized
    v4f r = ((const v4f*)(rel + (size_t)ri * HDIM))[lane];
    float ss = r.x * r.x + r.y * r.y + r.z * r.z + r.w * r.w;
    ss = wave_reduce_add(ss);
    const float invr = 1.0f / fmaxf(__builtin_sqrtf(ss), 1e-12f);
    ((v4f*)(query + (size_t)l * HDIM))[lane] = e + r * invr;
}

// ---- Kernel 3: dist[l][n] = sum_h |q[l][h] - e[n][h]| via WMMA ones-trick --
// A[m,k] = abs-diffs (16 entities x 4 h), B = ones -> D row-sums accumulate in C (f32 exact).
__global__ void __launch_bounds__(256) k_dist_wmma(const float* __restrict__ ent_norm,
                                                   const float* __restrict__ query,
                                                   float* __restrict__ dist) {
    __shared__ float e_lds[E_TILE * LDS_STRIDE];
    __shared__ float q_lds[Q_TILE * LDS_STRIDE];

    const int tid  = threadIdx.x;
    const int lane = tid & 31;
    const int wid  = tid >> 5;
    const int e0 = blockIdx.x * E_TILE;
    const int q0 = blockIdx.y * Q_TILE;

    // stage query tile (Q_TILE x 128 f32, float4 granularity)
    for (int idx = tid; idx < Q_TILE * 32; idx += 256) {
        const int row = idx >> 5, ch = idx & 31;
        async_cp16(query + (size_t)(q0 + row) * HDIM + ch * 4,
                   &q_lds[row * LDS_STRIDE + ch * 4]);
    }
    // stage entity tile (E_TILE x 128 f32), clamp OOB rows (stores are guarded later)
    for (int idx = tid; idx < E_TILE * 32; idx += 256) {
        const int row = idx >> 5, ch = idx & 31;
        int gr = e0 + row; if (gr >= N_NODES) gr = N_NODES - 1;
        async_cp16(ent_norm + (size_t)gr * HDIM + ch * 4,
                   &e_lds[row * LDS_STRIDE + ch * 4]);
    }
    async_wait_all();
    __syncthreads();

    // A layout (f32 16x4): lanes 0-15 -> M=lane, K=0,1 ; lanes 16-31 -> M=lane-16, K=2,3
    const int m  = lane & 15;
    const int ks = (lane >> 4) << 1;   // 0 or 2

    v2f ones; ones.x = 1.0f; ones.y = 1.0f;

    // 8 waves cover Q_TILE * (E_TILE/16) = 128 tasks
    for (int task = wid; task < Q_TILE * (E_TILE / 16); task += 8) {
        const int l_loc = task >> 2;
        const int et    = task & 3;
        const float* erow = &e_lds[(et * 16 + m) * LDS_STRIDE + ks];
        const float* qrow = &q_lds[l_loc * LDS_STRIDE + ks];

        v8f acc = {0.f, 0.f, 0.f, 0.f, 0.f, 0.f, 0.f, 0.f};
        #pragma unroll
        for (int c = 0; c < 32; ++c) {          // K = 128 in chunks of 4
            v2f e2 = *(const v2f*)(erow + 4 * c);
            v2f q2 = *(const v2f*)(qrow + 4 * c);
            v2f a;
            a.x = __builtin_fabsf(q2.x - e2.x);
            a.y = __builtin_fabsf(q2.y - e2.y);
            acc = __builtin_amdgcn_wmma_f32_16x16x4_f32(
                false, a, false, ones, (short)0, acc, false, false);
        }

        // D column 0: lane0 holds M=0..7 in acc[0..7], lane16 holds M=8..15
        const int l  = q0 + l_loc;
        const int nb = e0 + et * 16;
        if (lane == 0 || lane == 16) {
            const int mo = (lane == 0) ? 0 : 8;
            float* dst = dist + (size_t)l * N_NODES + nb + mo;
            if (nb + 15 < N_NODES) {
                #pragma unroll
                for (int i = 0; i < 8; ++i) dst[i] = acc[i];
            } else {
                for (int i = 0; i < 8; ++i)
                    if (nb + mo + i < N_NODES) dst[i] = acc[i];
            }
        }
    }
}

// ---- Kernel 4: softmax over each row of 10000 (in-place on d_out) ---------
__global__ void __launch_bounds__(256) k_softmax_rows(float* __restrict__ out) {
    __shared__ float sred[8];
    const int tid = threadIdx.x, lane = tid & 31, wid = tid >> 5;
    float* p = out + (size_t)blockIdx.x * N_NODES;

    float mx = -3.4e38f;
    for (int i = tid; i < N_NODES; i += 256) mx = fmaxf(mx, p[i]);
    #pragma unroll
    for (int off = 16; off > 0; off >>= 1) mx = fmaxf(mx, __shfl_xor(mx, off, 32));
    if (lane == 0) sred[wid] = mx;
    __syncthreads();
    if (wid == 0) {
        float t = sred[lane & 7];
        #pragma unroll
        for (int off = 4; off > 0; off >>= 1) t = fmaxf(t, __shfl_xor(t, off, 32));
        if (lane == 0) sred[0] = t;
    }
    __syncthreads();
    mx = sred[0];
    __syncthreads();   // everyone has read sred[0] before it is overwritten

    float s = 0.0f;
    for (int i = tid; i < N_NODES; i += 256) {
        const float e = expf(p[i] - mx);
        p[i] = e;
        s += e;
    }
    s = wave_reduce_add(s);
    if (lane == 0) sred[wid] = s;
    __syncthreads();
    if (wid == 0) {
        float t = sred[lane & 7];
        #pragma unroll
        for (int off = 4; off > 0; off >>= 1) t += __shfl_xor(t, off, 32);
        if (lane == 0) sred[0] = t;
    }
    __syncthreads();
    const float inv = 1.0f / sred[0];
    for (int i = tid; i < N_NODES; i += 256) p[i] *= inv;
}

extern "C" void kernel_launch(void* const* d_in, const int* in_sizes, int n_in,
                              void* d_out, int out_size, void* d_ws, size_t ws_size,
                              hipStream_t stream) {
    (void)in_sizes; (void)n_in; (void)out_size; (void)ws_size;
    const float* node = (const float*)d_in[0];
    const float* rel  = (const float*)d_in[1];
    const int*   eidx = (const int*)d_in[2];
    const int*   ridx = (const int*)d_in[3];
    float* out = (float*)d_out;

    float* ws_ent   = (float*)d_ws;                         // 10000*128 f32 = 5.12 MB
    float* ws_query = ws_ent + (size_t)N_NODES * HDIM;      // 128*128 f32 = 64 KB

    k_normalize_nodes<<<(N_NODES + 7) / 8, 256, 0, stream>>>(node, ws_ent);
    k_build_query<<<(LQ + 7) / 8, 256, 0, stream>>>(ws_ent, rel, eidx, ridx, ws_query);
    dim3 gridC((N_NODES + E_TILE - 1) / E_TILE, LQ / Q_TILE);
    k_dist_wmma<<<gridC, 256, 0, stream>>>(ws_ent, ws_query, out);
    k_softmax_rows<<<LQ, 256, 0, stream>>>(out);
}